// signalEncoding_72885595013453
// MI455X (gfx1250) — compile-verified
//
#include <hip/hip_runtime.h>
#include <hip/hip_bf16.h>

#define B_SZ  256
#define SDIM  512
#define NSIG  50

typedef __attribute__((ext_vector_type(2))) float v2f;
typedef __attribute__((ext_vector_type(8))) float v8f;

// reflect ('reflect' mode, no edge duplication) index into [0,512)
__device__ __forceinline__ int refl512(int p) {
    p = (p < 0) ? -p : p;
    return (p > 511) ? (1022 - p) : p;
}

__device__ __forceinline__ float selu_f(float x) {
    const float alpha = 1.6732632423543772f;
    const float scale = 1.0507009873554805f;
    return x > 0.0f ? scale * x : scale * alpha * (expf(x) - 1.0f);
}

// ---------------------------------------------------------------------------
// Kernel A: d = shared(shared(x)) + x ; LayerNorm over last axis -> d_norm
// One block per batch element. All intermediates in LDS.
// ---------------------------------------------------------------------------
__global__ __launch_bounds__(256) void shared_ln_kernel(
    const float* __restrict__ latent,
    const float* __restrict__ w1, const float* __restrict__ b1,
    const float* __restrict__ w2, const float* __restrict__ b2,
    const float* __restrict__ ln_g, const float* __restrict__ ln_b,
    float* __restrict__ d_norm)
{
    __shared__ float sx[4 * SDIM];     // input (kept for residual)
    __shared__ float sy[4 * SDIM];     // pass output / d
    __shared__ float sh[8 * SDIM];     // hidden (8 ch)
    __shared__ float sw1[160]; __shared__ float sb1[8];
    __shared__ float sw2[160]; __shared__ float sb2[4];
    __shared__ float red[256];
    __shared__ float stats[8];         // mu[4], rstd[4]

    const int b   = blockIdx.x;
    const int tid = threadIdx.x;

    // latent (B, 2048) -> (B, 512, 4) -> transpose -> sx[c][l] = latent[b, l*4+c]
    for (int i = tid; i < 2048; i += 256) {
        int c = i >> 9, l = i & 511;
        sx[c * SDIM + l] = latent[(size_t)b * 2048 + l * 4 + c];
    }
    for (int i = tid; i < 160; i += 256) { sw1[i] = w1[i]; sw2[i] = w2[i]; }
    if (tid < 8) sb1[tid] = b1[tid];
    if (tid < 4) sb2[tid] = b2[tid];
    __syncthreads();

    // two passes of sharedCNN: selu(conv 4->8 k5 d1 pad2) then conv 8->4 k5 d2 pad4
    for (int pass = 0; pass < 2; ++pass) {
        const float* in = (pass == 0) ? sx : sy;
        for (int i = tid; i < 8 * SDIM; i += 256) {
            int co = i >> 9, l = i & 511;
            float acc = sb1[co];
            #pragma unroll
            for (int ci = 0; ci < 4; ++ci)
                #pragma unroll
                for (int t = 0; t < 5; ++t)
                    acc += sw1[(co * 4 + ci) * 5 + t] * in[ci * SDIM + refl512(l - 2 + t)];
            sh[co * SDIM + l] = selu_f(acc);
        }
        __syncthreads();
        for (int i = tid; i < 4 * SDIM; i += 256) {
            int co = i >> 9, l = i & 511;
            float acc = sb2[co];
            #pragma unroll
            for (int ci = 0; ci < 8; ++ci)
                #pragma unroll
                for (int t = 0; t < 5; ++t)
                    acc += sw2[(co * 8 + ci) * 5 + t] * sh[ci * SDIM + refl512(l - 4 + 2 * t)];
            sy[co * SDIM + l] = acc;   // safe: conv stage reads only sh
        }
        __syncthreads();
    }

    // residual
    for (int i = tid; i < 2048; i += 256) sy[i] += sx[i];
    __syncthreads();

    // LayerNorm per (b, channel) over 512 positions, eps = 1e-10
    for (int c = 0; c < 4; ++c) {
        float v0 = sy[c * SDIM + tid], v1 = sy[c * SDIM + 256 + tid];
        red[tid] = v0 + v1;
        __syncthreads();
        for (int off = 128; off > 0; off >>= 1) {
            if (tid < off) red[tid] += red[tid + off];
            __syncthreads();
        }
        if (tid == 0) stats[c] = red[0] * (1.0f / 512.0f);
        __syncthreads();
        float mu = stats[c];
        float d0 = v0 - mu, d1 = v1 - mu;
        red[tid] = d0 * d0 + d1 * d1;
        __syncthreads();
        for (int off = 128; off > 0; off >>= 1) {
            if (tid < off) red[tid] += red[tid + off];
            __syncthreads();
        }
        if (tid == 0) stats[4 + c] = rsqrtf(red[0] * (1.0f / 512.0f) + 1e-10f);
        __syncthreads();
    }

    for (int c = 0; c < 4; ++c) {
        float mu = stats[c], rs = stats[4 + c];
        for (int l = tid; l < SDIM; l += 256)
            d_norm[((size_t)b * 4 + c) * SDIM + l] =
                (sy[c * SDIM + l] - mu) * rs * ln_g[l] + ln_b[l];
    }
}

// im2col B-element for conv1: k = ci*7 + tap (k < 28 always), reflect pad 3
__device__ __forceinline__ float im2col_b(const float* sx, int k, int n) {
    int ci  = k / 7;
    int tap = k - ci * 7;
    return sx[ci * SDIM + refl512(n - 3 + tap)];
}

// ---------------------------------------------------------------------------
// Kernel B: one block per (signal-PAIR, batch). Two signals share one WMMA
// tile: A rows 0-7 = sig s0 channels, rows 8-15 = sig s1 channels; the B
// matrix (im2col of d_norm) is identical for all signals, so M=16 is fully
// used. conv1 = 7 x V_WMMA_F32_16X16X4_F32 per 16-column tile (K=28).
// conv2/3/4 via VALU from LDS, per signal sequentially. Output (B, NSIG, 512).
// ---------------------------------------------------------------------------
__global__ __launch_bounds__(256) void signal_wmma_kernel(
    const float* __restrict__ d_norm,
    const float* __restrict__ w1, const float* __restrict__ b1,
    const float* __restrict__ w2, const float* __restrict__ b2,
    const float* __restrict__ w3, const float* __restrict__ b3,
    const float* __restrict__ w4, const float* __restrict__ b4,
    float* __restrict__ out)
{
    __shared__ float sx[4 * SDIM];          // 8 KB
    __shared__ float h1[2 * 8 * SDIM];      // 32 KB (both signals)
    __shared__ float h2[4 * SDIM];          // 8 KB  (reused per signal)
    __shared__ float h3[2 * SDIM];          // 4 KB  (reused per signal)
    __shared__ float sw1[2 * 224]; __shared__ float sb1[16];
    __shared__ float sw2[2 * 160]; __shared__ float sb2[8];
    __shared__ float sw3[2 * 40];  __shared__ float sb3[4];
    __shared__ float sw4[2 * 10];  __shared__ float sb4[2];

    const int pair = blockIdx.x;            // 0..24
    const int s0   = pair * 2;
    const int b    = blockIdx.y;
    const int tid  = threadIdx.x;
    const int lane = tid & 31;              // wave32
    const int wave = tid >> 5;              // 8 waves

    for (int i = tid; i < 2048; i += 256) sx[i]  = d_norm[(size_t)b * 2048 + i];
    for (int i = tid; i < 448;  i += 256) sw1[i] = w1[(size_t)s0 * 224 + i];
    for (int i = tid; i < 320;  i += 256) sw2[i] = w2[(size_t)s0 * 160 + i];
    if (tid < 80) sw3[tid] = w3[(size_t)s0 * 40 + tid];
    if (tid < 20) sw4[tid] = w4[(size_t)s0 * 10 + tid];
    if (tid < 16) sb1[tid] = b1[s0 * 8 + tid];
    if (tid < 8)  sb2[tid] = b2[s0 * 4 + tid];
    if (tid < 4)  sb3[tid] = b3[s0 * 2 + tid];
    if (tid < 2)  sb4[tid] = b4[s0 + tid];
    __syncthreads();

    // --- conv1 via V_WMMA_F32_16X16X4_F32, M fully packed ---------------
    // A (16x4 f32 layout): lanes 0-15 hold rows M=0..15; VGPR pair = K pair;
    // lane half selects K {0,1} vs {2,3}. Row m: signal (m>>3), channel (m&7).
    const int m     = lane & 15;
    const int khalf = (lane >> 4) << 1;        // 0 or 2
    const float* wrow = &sw1[(m >> 3) * 224 + (m & 7) * 28];
    v2f a[7];
    #pragma unroll
    for (int kg = 0; kg < 7; ++kg) {           // K = 28, no zero tail
        int k0 = kg * 4 + khalf;               // <= 27
        a[kg].x = wrow[k0];
        a[kg].y = wrow[k0 + 1];
    }

    for (int t = wave; t < 32; t += 8) {       // 32 N-tiles of 16 columns
        const int nb = t * 16;
        const int n  = nb + (lane & 15);
        // gather all B operands first so ds_loads batch ahead of the WMMA chain
        v2f bb[7];
        #pragma unroll
        for (int kg = 0; kg < 7; ++kg) {
            int k0 = kg * 4 + khalf;
            bb[kg].x = im2col_b(sx, k0,     n);
            bb[kg].y = im2col_b(sx, k0 + 1, n);
        }
        v8f acc;
        #pragma unroll
        for (int mm = 0; mm < 8; ++mm)         // bias: rows 0-7 sig0, 8-15 sig1
            acc[mm] = (lane < 16) ? sb1[mm] : sb1[8 + mm];
        #pragma unroll
        for (int kg = 0; kg < 7; ++kg)
            acc = __builtin_amdgcn_wmma_f32_16x16x4_f32(
                      false, a[kg], false, bb[kg], (short)0, acc, false, false);
        // D: VGPR mm, lanes 0-15 = row mm (sig0 ch mm), lanes 16-31 = row mm+8
        // (sig1 ch mm). Every lane stores -- no divergence.
        float* dst = &h1[(lane >> 4) * (8 * SDIM) + nb + (lane & 15)];
        #pragma unroll
        for (int mm = 0; mm < 8; ++mm)
            dst[mm * SDIM] = acc[mm];
    }
    __syncthreads();

    // --- conv2/3/4 per signal (VALU; tiny MAC counts) -------------------
    for (int sig = 0; sig < 2; ++sig) {
        const float* h1s = &h1[sig * 8 * SDIM];
        const float* w2s = &sw2[sig * 160]; const float* b2s = &sb2[sig * 4];
        const float* w3s = &sw3[sig * 40];  const float* b3s = &sb3[sig * 2];
        const float* w4s = &sw4[sig * 10];  const float  b4s = sb4[sig];

        // conv2: 8->4, k5 d1 pad2, selu
        for (int i = tid; i < 4 * SDIM; i += 256) {
            int co = i >> 9, l = i & 511;
            float acc = b2s[co];
            #pragma unroll
            for (int ci = 0; ci < 8; ++ci)
                #pragma unroll
                for (int t = 0; t < 5; ++t)
                    acc += w2s[(co * 8 + ci) * 5 + t] * h1s[ci * SDIM + refl512(l - 2 + t)];
            h2[i] = selu_f(acc);
        }
        __syncthreads();

        // conv3: 4->2, k5 d1 pad2
        for (int i = tid; i < 2 * SDIM; i += 256) {
            int co = i >> 9, l = i & 511;
            float acc = b3s[co];
            #pragma unroll
            for (int ci = 0; ci < 4; ++ci)
                #pragma unroll
                for (int t = 0; t < 5; ++t)
                    acc += w3s[(co * 4 + ci) * 5 + t] * h2[ci * SDIM + refl512(l - 2 + t)];
            h3[i] = acc;
        }
        __syncthreads();

        // conv4: 2->1, k5 d2 pad4, selu; write (B, NSIG, 512)
        for (int l = tid; l < SDIM; l += 256) {
            float acc = b4s;
            #pragma unroll
            for (int ci = 0; ci < 2; ++ci)
                #pragma unroll
                for (int t = 0; t < 5; ++t)
                    acc += w4s[ci * 5 + t] * h3[ci * SDIM + refl512(l - 4 + 2 * t)];
            out[((size_t)b * NSIG + s0 + sig) * SDIM + l] = selu_f(acc);
        }
        __syncthreads();   // h2/h3 reused by next signal
    }
}

extern "C" void kernel_launch(void* const* d_in, const int* in_sizes, int n_in,
                              void* d_out, int out_size, void* d_ws, size_t ws_size,
                              hipStream_t stream) {
    (void)in_sizes; (void)n_in; (void)out_size; (void)ws_size;
    const float* latent = (const float*)d_in[0];
    const float* sh_w1  = (const float*)d_in[1];
    const float* sh_b1  = (const float*)d_in[2];
    const float* sh_w2  = (const float*)d_in[3];
    const float* sh_b2  = (const float*)d_in[4];
    const float* ln_g   = (const float*)d_in[5];
    const float* ln_b   = (const float*)d_in[6];
    const float* sg_w1  = (const float*)d_in[7];
    const float* sg_b1  = (const float*)d_in[8];
    const float* sg_w2  = (const float*)d_in[9];
    const float* sg_b2  = (const float*)d_in[10];
    const float* sg_w3  = (const float*)d_in[11];
    const float* sg_b3  = (const float*)d_in[12];
    const float* sg_w4  = (const float*)d_in[13];
    const float* sg_b4  = (const float*)d_in[14];

    float* d_norm = (float*)d_ws;             // B*4*512 floats = 2 MB scratch
    float* outp   = (float*)d_out;            // (B, NSIG, 512)

    shared_ln_kernel<<<dim3(B_SZ), dim3(256), 0, stream>>>(
        latent, sh_w1, sh_b1, sh_w2, sh_b2, ln_g, ln_b, d_norm);

    signal_wmma_kernel<<<dim3(NSIG / 2, B_SZ), dim3(256), 0, stream>>>(
        d_norm, sg_w1, sg_b1, sg_w2, sg_b2, sg_w3, sg_b3, sg_w4, sg_b4, outp);
}